// KnnUnpoolLayer_86406152061327
// MI455X (gfx1250) — compile-verified
//
#include <hip/hip_runtime.h>
#include <hip/hip_bf16.h>

// CDNA5 / gfx1250 fused kNN-unpool.
// fp32 WMMA (V_WMMA_F32_16X16X4_F32) for the goals @ feats^T term, fused with
// on-the-fly top-3 selection and neighbor-mean scatter.
// One wave (32 lanes) owns a 32-goal stripe: two 16x64 A tiles -> two
// independent WMMA accumulator chains per 16-feature B tile (2-way ILP over
// the WMMA RAW chain, 2x B reuse), full-wave top-3 selection (lane == row).

typedef __attribute__((ext_vector_type(2))) float v2f;
typedef __attribute__((ext_vector_type(8))) float v8f;

#define DIM 64          // feature dimension
#define N0  4096        // initial rows
#define MG  32          // goals per wave (two 16-row WMMA tiles)

// ---------------------------------------------------------------- row norms
__global__ void row_norms_kernel(const float* __restrict__ feats,
                                 float* __restrict__ norms, int n) {
  int r = blockIdx.x * blockDim.x + threadIdx.x;
  if (r < n) {
    const float* row = feats + r * DIM;
    float s = 0.f;
#pragma unroll
    for (int i = 0; i < DIM; ++i) s += row[i] * row[i];
    norms[r] = s;
  }
}

// --------------------------------------------- deterministic permutation sub
// odd multiplier mod 2^13 is a bijection on [0,8192): stand-in for
// jax.random.permutation(key(42), 8192)[:4096]
__global__ void make_goal_idx_kernel(int* __restrict__ idx, int g, unsigned nmask) {
  int i = blockIdx.x * blockDim.x + threadIdx.x;
  if (i < g) idx[i] = (int)((2654435761u * (unsigned)i) & nmask);
}

// ----------------------------------------------------------- fused kNN stage
__global__ __launch_bounds__(32)
void knn_wmma_kernel(const float* __restrict__ feats,    // N x 64
                     const float* __restrict__ norms,    // N
                     const int*   __restrict__ goal_idx, // G entries or null
                     int N,
                     float* __restrict__ out_new)        // G x 64
{
  __shared__ float s_d[MG * 16];   // two stacked 16x16 score tiles
  __shared__ int   s_idx[MG * 3];

  const int lane  = threadIdx.x;        // 0..31
  const int gbase = blockIdx.x * MG;
  const int mh    = lane & 15;          // M (A) / N (B,D) position
  const int kh    = (lane >> 4) * 2;    // K sub-offset: lanes 16..31 hold K=2,3

  // Two A tiles: goal rows [gbase, gbase+16) and [gbase+16, gbase+32).
  int g0 = gbase + mh;
  int g1 = gbase + 16 + mh;
  if (goal_idx) { g0 = goal_idx[g0]; g1 = goal_idx[g1]; }
  const float* gp0 = feats + g0 * DIM;
  const float* gp1 = feats + g1 * DIM;

  v2f a0[16], a1[16];
#pragma unroll
  for (int kk = 0; kk < 16; ++kk) {
    a0[kk].x = gp0[kk * 4 + kh + 0];
    a0[kk].y = gp0[kk * 4 + kh + 1];
    a1[kk].x = gp1[kk * 4 + kh + 0];
    a1[kk].y = gp1[kk * 4 + kh + 1];
  }

  // lane l owns goal row (gbase + l): running top-3 smallest scores
  float best0 = 3.4e38f, best1 = 3.4e38f, best2 = 3.4e38f;
  int   i0 = 0, i1 = 0, i2 = 0;

  for (int fbase = 0; fbase < N; fbase += 16) {
    // Batch-load all 16 B fragments first so the loads pipeline together.
    const float* fp = feats + (fbase + mh) * DIM;
    v2f b[16];
#pragma unroll
    for (int kk = 0; kk < 16; ++kk) {
      b[kk].x = fp[kk * 4 + kh + 0];
      b[kk].y = fp[kk * 4 + kh + 1];
    }
    const float fn = norms[fbase + mh];

    // Two independent WMMA chains share each B fragment.
    v8f acc0 = {}, acc1 = {};
#pragma unroll
    for (int kk = 0; kk < 16; ++kk) {
      acc0 = __builtin_amdgcn_wmma_f32_16x16x4_f32(
          false, a0[kk], false, b[kk], (short)0, acc0, false, false);
      acc1 = __builtin_amdgcn_wmma_f32_16x16x4_f32(
          false, a1[kk], false, b[kk], (short)0, acc1, false, false);
    }

    // rank score: ||f||^2 - 2*dot  (||g||^2 is constant per goal row)
    const int mbase = (lane >> 4) << 3;   // rows 0..7 / 8..15 of each tile
#pragma unroll
    for (int r = 0; r < 8; ++r) {
      s_d[(mbase + r) * 16 + mh]        = fn - 2.0f * acc0[r];
      s_d[(16 + mbase + r) * 16 + mh]   = fn - 2.0f * acc1[r];
    }
    __syncthreads();

    // all 32 lanes: scan own row's 16 candidates; strict '<' + ascending
    // index preserves the reference's lowest-index tie-break (incl. self).
#pragma unroll
    for (int j = 0; j < 16; ++j) {
      float d  = s_d[lane * 16 + j];
      int   fi = fbase + j;
      if (d < best0)      { best2=best1; i2=i1; best1=best0; i1=i0; best0=d; i0=fi; }
      else if (d < best1) { best2=best1; i2=i1; best1=d;     i1=fi; }
      else if (d < best2) { best2=d;     i2=fi; }
    }
    __syncthreads();
  }

  s_idx[lane * 3 + 0] = i0;
  s_idx[lane * 3 + 1] = i1;
  s_idx[lane * 3 + 2] = i2;
  __syncthreads();

  // neighbor mean: 32 rows x 64 dims, 32 lanes cooperative
  const float inv3 = 1.0f / 3.0f;
  for (int t = lane; t < MG * DIM; t += 32) {
    int row = t >> 6, d = t & 63;
    int j0 = s_idx[row * 3 + 0];
    int j1 = s_idx[row * 3 + 1];
    int j2 = s_idx[row * 3 + 2];
    out_new[(gbase + row) * DIM + d] =
        (feats[j0 * DIM + d] + feats[j1 * DIM + d] + feats[j2 * DIM + d]) * inv3;
  }
}

// ---------------------------------------------------------------------------
extern "C" void kernel_launch(void* const* d_in, const int* in_sizes, int n_in,
                              void* d_out, int out_size, void* d_ws, size_t ws_size,
                              hipStream_t stream) {
  (void)in_sizes; (void)n_in; (void)out_size; (void)ws_size;

  const float* x   = (const float*)d_in[0];
  float*       out = (float*)d_out;

  float* norms = (float*)d_ws;                                  // 8192 f32
  int*   gidx  = (int*)((char*)d_ws + 8192 * sizeof(float));    // 4096 i32

  // rows [0, 4096) = x
  hipMemcpyAsync(out, x, (size_t)N0 * DIM * sizeof(float),
                 hipMemcpyDeviceToDevice, stream);

  // norms of rows [0, 4096)
  row_norms_kernel<<<N0 / 64, 64, 0, stream>>>(out, norms, N0);

  // phase 1: goals = feats (identity), N = 4096 -> writes rows [4096, 8192)
  knn_wmma_kernel<<<N0 / MG, 32, 0, stream>>>(out, norms, nullptr, N0,
                                              out + N0 * DIM);

  // norms of new rows [4096, 8192)
  row_norms_kernel<<<N0 / 64, 64, 0, stream>>>(out + N0 * DIM, norms + N0, N0);

  // deterministic permutation stand-in for goal selection
  make_goal_idx_kernel<<<N0 / 64, 64, 0, stream>>>(gidx, N0, 8191u);

  // phase 2: feats = rows [0, 8192), 4096 permuted goals -> rows [8192, 12288)
  knn_wmma_kernel<<<N0 / MG, 32, 0, stream>>>(out, norms, gidx, 2 * N0,
                                              out + 2 * N0 * DIM);
}